// CurvatureLoss_17428977287911
// MI455X (gfx1250) — compile-verified
//
#include <hip/hip_runtime.h>

typedef __attribute__((ext_vector_type(2))) float v2f;
typedef __attribute__((ext_vector_type(8))) float v8f;

// Exact fp32 full-wave32 sum using V_WMMA_F32_16X16X4_F32.
// A(16x4): lane L holds M=L%16; one K slot = value, other = 0.
// B(4x16) = all ones -> D[m][n] = acc(lane m) + acc(lane m+16).
// Lanes 0-15 see D rows 0..7 in d[0..7]; lanes 16-31 see rows 8..15.
// Sum the 8 D registers, then fold halves with shfl_xor(16).
__device__ __forceinline__ float wave32_sum(float x) {
    v2f a; a[0] = x;    a[1] = 0.0f;
    v2f b; b[0] = 1.0f; b[1] = 1.0f;
    v8f c = {};
    v8f d = __builtin_amdgcn_wmma_f32_16x16x4_f32(
        /*neg_a=*/false, a, /*neg_b=*/false, b,
        /*c_mod=*/(short)0, c, /*reuse_a=*/false, /*reuse_b=*/false);
    float s = d[0] + d[1] + d[2] + d[3] + d[4] + d[5] + d[6] + d[7];
    s += __shfl_xor(s, 16, 32);
    return s;  // full wave sum, uniform across lanes
}

struct Row8 { float v[8]; float l, r; };

// Branchless (per-lane) halo: always load from a clamped in-bounds address,
// scale by 0/1 weight. Zero-padding semantics preserved exactly.
// The y-bounds check is wave-uniform -> cheap scalar branch.
__device__ __forceinline__ Row8 load_diff_row8(const float* __restrict__ p,
                                               const float* __restrict__ t,
                                               int y, int x0, int xl, int xr,
                                               float lmul, float rmul,
                                               int H, int W) {
    Row8 r;
    if (y < 0 || y >= H) {
#pragma unroll
        for (int j = 0; j < 8; ++j) r.v[j] = 0.f;
        r.l = 0.f; r.r = 0.f;
        return r;
    }
    const float* pr = p + (size_t)y * W;
    const float* tr = t + (size_t)y * W;
    float4 pa = *reinterpret_cast<const float4*>(pr + x0);
    float4 pb = *reinterpret_cast<const float4*>(pr + x0 + 4);
    float4 ta = *reinterpret_cast<const float4*>(tr + x0);
    float4 tb = *reinterpret_cast<const float4*>(tr + x0 + 4);
    r.v[0] = pa.x - ta.x; r.v[1] = pa.y - ta.y;
    r.v[2] = pa.z - ta.z; r.v[3] = pa.w - ta.w;
    r.v[4] = pb.x - tb.x; r.v[5] = pb.y - tb.y;
    r.v[6] = pb.z - tb.z; r.v[7] = pb.w - tb.w;
    r.l = (pr[xl] - tr[xl]) * lmul;   // unconditional load, weight 0 at edge
    r.r = (pr[xr] - tr[xr]) * rmul;
    return r;
}

#define ROWS 16          // rows per thread
#define CHUNK 32         // rows per block (2 half-chunks of 16)

__global__ __launch_bounds__(256) void curv_main(
    const float* __restrict__ pred, const float* __restrict__ target,
    const float* __restrict__ mask, float* __restrict__ partNum,
    float* __restrict__ partDen) {
    constexpr int H = 1024, W = 1024;
    const int chunksPerImg = H / CHUNK;                 // 32
    const int blk   = blockIdx.x;
    const int img   = blk / chunksPerImg;
    const int chunk = blk % chunksPerImg;

    const int col  = threadIdx.x & 127;                 // 128 cols * 8 = W
    const int half = threadIdx.x >> 7;                  // 0 or 1
    const int x0   = col * 8;
    const bool hasL = (col > 0);
    const bool hasR = (col < 127);
    const int   xl   = hasL ? x0 - 1 : x0;
    const int   xr   = hasR ? x0 + 8 : x0 + 7;
    const float lmul = hasL ? 1.f : 0.f;
    const float rmul = hasR ? 1.f : 0.f;

    const size_t off = (size_t)img * H * W;
    const float* p = pred + off;
    const float* t = target + off;
    const float* m = mask + off;

    const int y0 = chunk * CHUNK + half * ROWS;
    Row8 up = load_diff_row8(p, t, y0 - 1, x0, xl, xr, lmul, rmul, H, W);
    Row8 ce = load_diff_row8(p, t, y0,     x0, xl, xr, lmul, rmul, H, W);

    float num = 0.f, den = 0.f;
#pragma unroll 4
    for (int i = 0; i < ROWS; ++i) {
        const int y = y0 + i;
        Row8 dn = load_diff_row8(p, t, y + 1, x0, xl, xr, lmul, rmul, H, W);
        const float* mr = m + (size_t)y * W + x0;
        float4 ma = *reinterpret_cast<const float4*>(mr);
        float4 mb = *reinterpret_cast<const float4*>(mr + 4);
        float mv[8] = { ma.x, ma.y, ma.z, ma.w, mb.x, mb.y, mb.z, mb.w };

        // 5-point Laplacian of (pred - target), SAME zero padding.
#pragma unroll
        for (int j = 0; j < 8; ++j) {
            float left  = (j == 0) ? ce.l : ce.v[j - 1];
            float right = (j == 7) ? ce.r : ce.v[j + 1];
            float lap   = up.v[j] + dn.v[j] + left + right - 4.f * ce.v[j];
            num += fabsf(lap) * mv[j];
            den += mv[j];
        }
        up = ce; ce = dn;
    }

    // Wave reduction via WMMA (EXEC all ones here: no divergence yet).
    float wnum = wave32_sum(num);
    float wden = wave32_sum(den);

    __shared__ float sn[8], sd[8];
    const int wid = threadIdx.x >> 5;
    if ((threadIdx.x & 31) == 0) { sn[wid] = wnum; sd[wid] = wden; }
    __syncthreads();
    if (threadIdx.x == 0) {
        float a = 0.f, d = 0.f;
#pragma unroll
        for (int i = 0; i < 8; ++i) { a += sn[i]; d += sd[i]; }
        partNum[blk] = a;  // deterministic: one writer per slot, no atomics
        partDen[blk] = d;
    }
}

__global__ __launch_bounds__(256) void curv_final(
    const float* __restrict__ partNum, const float* __restrict__ partDen,
    int nPart, float* __restrict__ out) {
    float num = 0.f, den = 0.f;
    for (int i = threadIdx.x; i < nPart; i += 256) {  // uniform trip count
        num += partNum[i];
        den += partDen[i];
    }
    num = wave32_sum(num);
    den = wave32_sum(den);

    __shared__ float sn[8], sd[8];
    const int wid = threadIdx.x >> 5;
    if ((threadIdx.x & 31) == 0) { sn[wid] = num; sd[wid] = den; }
    __syncthreads();
    if (threadIdx.x == 0) {
        float a = 0.f, d = 0.f;
#pragma unroll
        for (int i = 0; i < 8; ++i) { a += sn[i]; d += sd[i]; }
        out[0] = a / (d + 1e-8f);
    }
}

extern "C" void kernel_launch(void* const* d_in, const int* in_sizes, int n_in,
                              void* d_out, int out_size, void* d_ws, size_t ws_size,
                              hipStream_t stream) {
    const float* pred   = (const float*)d_in[0];
    const float* target = (const float*)d_in[1];
    const float* mask   = (const float*)d_in[2];
    float* out = (float*)d_out;

    constexpr int H = 1024, W = 1024;
    const int nimg    = in_sizes[0] / (H * W);        // 16
    const int nBlocks = nimg * (H / CHUNK);           // 512 blocks x 256 thr

    float* partNum = (float*)d_ws;                    // nBlocks floats
    float* partDen = partNum + nBlocks;               // nBlocks floats (2 KB)

    curv_main<<<nBlocks, 256, 0, stream>>>(pred, target, mask, partNum, partDen);
    curv_final<<<1, 256, 0, stream>>>(partNum, partDen, nBlocks, out);
}